// PoolingNms1_22110491640314
// MI455X (gfx1250) — compile-verified
//
#include <hip/hip_runtime.h>
#include <cstdint>
#include <cstddef>

// Problem constants (fixed by the reference setup): x is (16,1,1536,1536) f32, k=3.
#define IMG_N     16
#define IMG_W     1536
#define KPOOL     3
#define BLK_ROWS  (IMG_W / KPOOL)          // 512 block-rows per image
#define STRIP_F   (KPOOL * IMG_W)          // 4608 floats per 3-row strip (contiguous)
#define STRIP_V4  (STRIP_F / 4)            // 1152 float4 (16B) chunks per strip
#define NBLK      (IMG_W / KPOOL)          // 512 blocks across a strip
#define NTHREADS  256

__global__ __launch_bounds__(NTHREADS)
void pool_nms_kernel(const float* __restrict__ x, float* __restrict__ out) {
    // 18 KB LDS staging buffer for one 3-row strip.
    __shared__ float lds[STRIP_F];

    const uint32_t tid  = threadIdx.x;
    const size_t   strip_base = (size_t)blockIdx.x * STRIP_F;
    const float*   gsrc = x + strip_base;

    // ---- Phase 1: async copy strip (contiguous 18432 B) global -> LDS ----
    // CDNA5 async tensor-class path: global_load_async_to_lds_b128, ASYNCcnt.
    const uint32_t lds_base = (uint32_t)(uintptr_t)(&lds[0]);
    for (uint32_t i = tid; i < STRIP_V4; i += NTHREADS) {
        const uint32_t goff  = i * 16u;           // byte offset within strip
        const uint32_t laddr = lds_base + goff;   // LDS byte address
        asm volatile("global_load_async_to_lds_b128 %0, %1, %2"
                     :
                     : "v"(laddr), "v"(goff), "s"(gsrc)
                     : "memory");
    }
    asm volatile("s_wait_asynccnt 0" ::: "memory");
    __syncthreads();

    // ---- Phase 2: per-block first-argmax, scatter value / zeros ----
    float* gdst = out + strip_base;
    for (uint32_t j = tid; j < NBLK; j += NTHREADS) {
        const uint32_t c0 = j * KPOOL;

        float v[KPOOL * KPOOL];
        #pragma unroll
        for (int r = 0; r < KPOOL; ++r) {
            #pragma unroll
            for (int c = 0; c < KPOOL; ++c) {
                v[r * KPOOL + c] = lds[r * IMG_W + c0 + c];
            }
        }

        // First occurrence of the maximum in (dy,dx) row-major order
        // (strict '>' keeps the earliest max, matching jnp.argmax).
        int   best = 0;
        float bv   = v[0];
        #pragma unroll
        for (int i = 1; i < KPOOL * KPOOL; ++i) {
            if (v[i] > bv) { bv = v[i]; best = i; }
        }

        // Streaming (non-temporal) stores: write-once output, no reuse,
        // keep it out of L2 (working set 302 MB > 192 MB L2).
        #pragma unroll
        for (int r = 0; r < KPOOL; ++r) {
            #pragma unroll
            for (int c = 0; c < KPOOL; ++c) {
                const int   idx = r * KPOOL + c;
                const float o   = (idx == best) ? v[idx] : 0.0f;
                __builtin_nontemporal_store(o, &gdst[r * IMG_W + c0 + c]);
            }
        }
    }
}

extern "C" void kernel_launch(void* const* d_in, const int* in_sizes, int n_in,
                              void* d_out, int out_size, void* d_ws, size_t ws_size,
                              hipStream_t stream) {
    const float* x   = (const float*)d_in[0];
    float*       out = (float*)d_out;
    // 16 images * 512 block-rows = 8192 strips, one workgroup each.
    const int strips = IMG_N * BLK_ROWS;
    hipLaunchKernelGGL(pool_nms_kernel, dim3(strips), dim3(NTHREADS), 0, stream,
                       x, out);
}